// GCNModel_86844238725155
// MI455X (gfx1250) — compile-verified
//
#include <hip/hip_runtime.h>
#include <hip/hip_bf16.h>

// ---------------------------------------------------------------------------
// 3-layer GCN (PyG GCNConv semantics) for MI455X / gfx1250.
//   deg/norm precomputed once; per layer: GEMM (WMMA f32 16x16x4 for the
//   64-wide layers) with the self-loop init (+bias) fused into the epilogue,
//   edge scatter via hardware global_atomic_add_f32, relu fused into the
//   next GEMM's A load. Entire pipeline is HBM/L2-bandwidth bound; GEMM
//   FLOPs (~1 GFLOP total) are noise, so fp32 WMMA keeps full precision.
// ---------------------------------------------------------------------------

#define N_NODES 100000
#define F_HID   64

typedef __attribute__((ext_vector_type(2))) float v2f;
typedef __attribute__((ext_vector_type(8))) float v8f;

__device__ __forceinline__ void atomAddF(float* p, float v) {
  // hardware global_atomic_add_f32 (no CAS loop)
  unsafeAtomicAdd(p, v);
}

// ---------------- graph normalization ----------------

__global__ void k_fill1(float* deg, int n) {
  int i = blockIdx.x * blockDim.x + threadIdx.x;
  if (i < n) deg[i] = 1.0f;  // self-loop contributes 1
}

__global__ void k_deg(const int* __restrict__ dst, float* deg, int e) {
  int i = blockIdx.x * blockDim.x + threadIdx.x;
  if (i < e) atomAddF(&deg[dst[i]], 1.0f);
}

__global__ void k_dinv(float* deg, float* __restrict__ nself, int n) {
  int i = blockIdx.x * blockDim.x + threadIdx.x;
  if (i >= n) return;
  float d = deg[i];
  float r = (d > 0.0f) ? rsqrtf(d) : 0.0f;
  deg[i] = r;            // dinv in place
  nself[i] = r * r;      // self-loop norm
}

__global__ void k_norme(const int* __restrict__ src, const int* __restrict__ dst,
                        const float* __restrict__ dinv, float* __restrict__ norme, int e) {
  int i = blockIdx.x * blockDim.x + threadIdx.x;
  if (i < e) norme[i] = dinv[src[i]] * dinv[dst[i]];
}

// ---------------- layer 1 GEMM: [N,3] @ [3,64], fused self-init ------------

__global__ void k_gemm1(const float* __restrict__ x, const float* __restrict__ W1,
                        const float* __restrict__ nself, const float* __restrict__ bias,
                        float* __restrict__ hA, float* __restrict__ hB) {
  int i = blockIdx.x * blockDim.x + threadIdx.x;       // N*32 threads
  if (i >= N_NODES * 32) return;
  int n  = i >> 5;
  int f2 = (i & 31) * 2;
  float x0 = x[n * 3 + 0], x1 = x[n * 3 + 1], x2 = x[n * 3 + 2];
  float2 w0 = *(const float2*)(W1 + 0 * 64 + f2);
  float2 w1 = *(const float2*)(W1 + 1 * 64 + f2);
  float2 w2 = *(const float2*)(W1 + 2 * 64 + f2);
  float2 o;
  o.x = fmaf(x0, w0.x, fmaf(x1, w1.x, x2 * w2.x));
  o.y = fmaf(x0, w0.y, fmaf(x1, w1.y, x2 * w2.y));
  *(float2*)(hA + (size_t)n * 64 + f2) = o;            // scatter source
  float ns = nself[n];
  float2 b = *(const float2*)(bias + f2);
  float2 s;
  s.x = fmaf(o.x, ns, b.x);
  s.y = fmaf(o.y, ns, b.y);
  *(float2*)(hB + (size_t)n * 64 + f2) = s;            // aggregation init (+bias)
}

// ---------------- edge scatter, 64 features: 1 wave32 per edge -------------

__global__ void __launch_bounds__(256)
k_scatter64(const float* __restrict__ h, const int* __restrict__ src,
            const int* __restrict__ dst, const float* __restrict__ norme,
            float* out, int e) {
  int edge = blockIdx.x * (blockDim.x >> 5) + (threadIdx.x >> 5);
  int lane = threadIdx.x & 31;
  if (edge >= e) return;
  int s = src[edge], d = dst[edge];
  float w = norme[edge];
  float2 v = *(const float2*)(h + (size_t)s * 64 + lane * 2);
  float* o = out + (size_t)d * 64 + lane * 2;
  atomAddF(o + 0, v.x * w);
  atomAddF(o + 1, v.y * w);
}

// ---------------- layer 2 GEMM: relu(hin)[N,64] @ W[64,64] via WMMA --------
// 64 rows per block (4 row-tiles per wave) to amortize W staging in LDS.
// Each wave owns one 16-col tile; K=64 -> 16 x V_WMMA_F32_16X16X4_F32.
// Epilogue stores both hA (scatter source) and hB = acc*nself[row]+bias[col].

#define LPAD 66
#define G2ROWS 64

__global__ void __launch_bounds__(128)
k_gemm2(const float* __restrict__ hin, const float* __restrict__ W,
        const float* __restrict__ nself, const float* __restrict__ bias,
        float* __restrict__ hA, float* __restrict__ hB, int n_nodes) {
  __shared__ __align__(16) float sA[G2ROWS * LPAD];   // A tile, relu applied
  __shared__ __align__(16) float sB[64 * LPAD];       // sB[n][k] = W[k][n]
  int tid = threadIdx.x;
  int rowBase = blockIdx.x * G2ROWS;

  for (int i = tid; i < G2ROWS * 64; i += 128) {
    int r = i >> 6, c = i & 63;
    int row = rowBase + r;
    float v = (row < n_nodes) ? fmaxf(hin[(size_t)row * 64 + c], 0.0f) : 0.0f;
    sA[r * LPAD + c] = v;
  }
  for (int i = tid; i < 64 * 64; i += 128) {
    int k = i >> 6, n = i & 63;
    sB[n * LPAD + k] = W[i];
  }
  __syncthreads();

  int wave = tid >> 5, lane = tid & 31;
  int m   = lane & 15;
  int khi = (lane >> 4) * 2;       // lanes 16-31 carry K = k0+2, k0+3
  int col = wave * 16 + m;         // this wave's output column (= sB row)
  float bcol = bias[col];
  int rowOff = (lane >> 4) * 8;    // C/D: VGPR r -> (M = r + 8*hi, N = lane&15)

  for (int rt = 0; rt < G2ROWS / 16; ++rt) {
    int tileRow = rowBase + rt * 16;
    if (tileRow >= n_nodes) break;               // wave-uniform, 16 | n_nodes

    v8f acc = {0.f, 0.f, 0.f, 0.f, 0.f, 0.f, 0.f, 0.f};
#pragma unroll
    for (int k0 = 0; k0 < 64; k0 += 4) {
      v2f a = *(const v2f*)(sA + (rt * 16 + m) * LPAD + k0 + khi);
      v2f b = *(const v2f*)(sB + col * LPAD + k0 + khi);
      acc = __builtin_amdgcn_wmma_f32_16x16x4_f32(false, a, false, b,
                                                  (short)0, acc, false, false);
    }

#pragma unroll
    for (int r = 0; r < 8; ++r) {
      int row = tileRow + rowOff + r;
      float v = acc[r];
      hA[(size_t)row * 64 + col] = v;                        // scatter source
      hB[(size_t)row * 64 + col] = fmaf(v, nself[row], bcol); // agg init (+bias)
    }
  }
}

// ---------------- layer 3 GEMM: relu(hin)[N,64] @ W3[64,3] via WMMA --------
// One 16-row tile per wave (cols >=3 zero-padded in B); epilogue also writes
// the aggregation init (+b3) straight into d_out.

__global__ void __launch_bounds__(128)
k_gemm3(const float* __restrict__ hin, const float* __restrict__ W3,
        const float* __restrict__ nself, const float* __restrict__ b3,
        float* __restrict__ h3, float* __restrict__ out) {
  int wave = blockIdx.x * 4 + (threadIdx.x >> 5);
  if (wave >= N_NODES / 16) return;               // wave-uniform: EXEC stays full
  int lane = threadIdx.x & 31;
  int m   = lane & 15;
  int khi = (lane >> 4) * 2;
  int rowBase = wave * 16;

  v8f acc = {0.f, 0.f, 0.f, 0.f, 0.f, 0.f, 0.f, 0.f};
#pragma unroll
  for (int k0 = 0; k0 < 64; k0 += 4) {
    float2 av = *(const float2*)(hin + (size_t)(rowBase + m) * 64 + k0 + khi);
    v2f a; a.x = fmaxf(av.x, 0.0f); a.y = fmaxf(av.y, 0.0f);
    v2f b;
    if (m < 3) { b.x = W3[(k0 + khi) * 3 + m]; b.y = W3[(k0 + khi + 1) * 3 + m]; }
    else       { b.x = 0.0f;                   b.y = 0.0f; }
    acc = __builtin_amdgcn_wmma_f32_16x16x4_f32(false, a, false, b,
                                                (short)0, acc, false, false);
  }
  if (m < 3) {
    int rowOff = (lane >> 4) * 8;
    float bm = b3[m];
#pragma unroll
    for (int r = 0; r < 8; ++r) {
      int row = rowBase + rowOff + r;
      float v = acc[r];
      h3[(size_t)row * 3 + m]  = v;                      // scatter source
      out[(size_t)row * 3 + m] = fmaf(v, nself[row], bm); // agg init (+bias)
    }
  }
}

// ---------------- 3-wide edge scatter --------------------------------------

__global__ void k_scatter3(const float* __restrict__ h3, const int* __restrict__ src,
                           const int* __restrict__ dst, const float* __restrict__ norme,
                           float* out, int e) {
  int i = blockIdx.x * blockDim.x + threadIdx.x;
  if (i >= e) return;
  int s = src[i], d = dst[i];
  float w = norme[i];
  atomAddF(&out[d * 3 + 0], h3[s * 3 + 0] * w);
  atomAddF(&out[d * 3 + 1], h3[s * 3 + 1] * w);
  atomAddF(&out[d * 3 + 2], h3[s * 3 + 2] * w);
}

// ---------------------------------------------------------------------------

extern "C" void kernel_launch(void* const* d_in, const int* in_sizes, int n_in,
                              void* d_out, int out_size, void* d_ws, size_t ws_size,
                              hipStream_t stream) {
  const float* x  = (const float*)d_in[0];
  const int*   ei = (const int*)d_in[1];
  const float* W1 = (const float*)d_in[2];
  const float* b1 = (const float*)d_in[3];
  const float* W2 = (const float*)d_in[4];
  const float* b2 = (const float*)d_in[5];
  const float* W3 = (const float*)d_in[6];
  const float* b3 = (const float*)d_in[7];
  float* out = (float*)d_out;

  const int N = N_NODES;
  const int E = in_sizes[1] / 2;      // 1,600,000
  const int* src = ei;
  const int* dst = ei + E;

  // workspace carve-up (all 8B-aligned): ~58.4 MB total
  float* dinv  = (float*)d_ws;                 // N   (degree -> dinv in place)
  float* nself = dinv  + N;                    // N
  float* norme = nself + N;                    // E
  float* hA    = norme + E;                    // N*64
  float* hB    = hA + (size_t)N * F_HID;       // N*64

  const int T = 256;
  // --- normalization (computed once, shared by all 3 layers) ---
  k_fill1<<<(N + T - 1) / T, T, 0, stream>>>(dinv, N);
  k_deg  <<<(E + T - 1) / T, T, 0, stream>>>(dst, dinv, E);
  k_dinv <<<(N + T - 1) / T, T, 0, stream>>>(dinv, nself, N);
  k_norme<<<(E + T - 1) / T, T, 0, stream>>>(src, dst, dinv, norme, E);

  // --- layer 1: hA = x@W1 ; hB = hA*nself + b1 (fused) ; scatter edges ---
  k_gemm1    <<<(N * 32 + T - 1) / T, T, 0, stream>>>(x, W1, nself, b1, hA, hB);
  k_scatter64<<<(E + 7) / 8, T, 0, stream>>>(hA, src, dst, norme, hB, E);

  // --- layer 2: hA = relu(hB)@W2 ; hB = hA*nself + b2 (fused) ; scatter ---
  k_gemm2    <<<(N + G2ROWS - 1) / G2ROWS, 128, 0, stream>>>(hB, W2, nself, b2,
                                                             hA, hB, N);
  k_scatter64<<<(E + 7) / 8, T, 0, stream>>>(hA, src, dst, norme, hB, E);

  // --- layer 3: h3 = relu(hB)@W3 -> hA ; out = h3*nself + b3 (fused) ------
  k_gemm3    <<<(N / 16 + 3) / 4, 128, 0, stream>>>(hB, W3, nself, b3, hA, out);
  k_scatter3 <<<(E + T - 1) / T, T, 0, stream>>>(hA, src, dst, norme, out, E);
}